// LogLikelihoodLayer_27204322852872
// MI455X (gfx1250) — compile-verified
//
#include <hip/hip_runtime.h>

typedef float v2f __attribute__((ext_vector_type(2)));
typedef float v8f __attribute__((ext_vector_type(8)));

#define SLDA 68                       // 64 + 4 pad: column accesses stride 4 banks/row
#define LBUF (64 * SLDA)              // floats per sigma tile buffer (17408 B)
#define LOG2PI 1.8378770664093453f
#define NPROB (64 * 400)
#define PPB 2                         // problems per block (double-buffered)

// Generic pointer -> raw 32-bit LDS offset (addrspace(3) pointers are LDS offsets)
__device__ __forceinline__ unsigned lds_offset(const void* p) {
  return (unsigned)(unsigned long long)(const __attribute__((address_space(3))) char*)p;
}

// Issue 32 per-lane async B128 copies: sigma tile (64x64 f32) -> padded LDS.
// Non-temporal: sigma is 419MB streamed once, don't let it thrash the 192MB L2.
__device__ __forceinline__ void async_load_sigma(float* Sbuf, const float* sig, int l) {
#pragma unroll
  for (int it = 0; it < 32; ++it) {
    int e = it * 128 + l * 4;                     // linear f32 element index
    int r = e >> 6, c = e & 63;
    unsigned loff = lds_offset(&Sbuf[r * SLDA + c]);
    const float* gp = sig + e;
    asm volatile("global_load_async_to_lds_b128 %0, %1, off th:TH_LOAD_NT"
                 :: "v"(loff), "v"(gp) : "memory");
  }
}

// Blocked Cholesky + forward solve + ll for one 64x64 problem resident in LDS.
__device__ __forceinline__ void chol_ll_one(float* S, const float* __restrict__ x,
                                            const float* __restrict__ mu, int pid,
                                            float* __restrict__ ll_out, int l) {
  const int m = l & 15;               // WMMA lane row/col index
  const int h = l >> 4;               // WMMA half-wave select

  // ---- blocked Cholesky, 4 blocks of 16 ----
  for (int k = 0; k < 4; ++k) {
    const int bk = 16 * k;

    // (1) factor 16x16 diagonal block in LDS (lanes 0..15 own rows)
    for (int j = 0; j < 16; ++j) {
      float d = S[(bk + j) * SLDA + (bk + j)];
      float s = sqrtf(d);
      float inv = 1.0f / s;
      if (l == j) S[(bk + j) * SLDA + (bk + j)] = s;
      else if (l < 16 && l > j) S[(bk + l) * SLDA + (bk + j)] *= inv;
      __syncthreads();
      if (l < 16 && l > j) {
        float lij = S[(bk + l) * SLDA + (bk + j)];
        for (int c = j + 1; c <= l; ++c)
          S[(bk + l) * SLDA + (bk + c)] -= lij * S[(bk + c) * SLDA + (bk + j)];
      }
      __syncthreads();
    }

    // (2) panel triangular solves: L_ik = A_ik * L_kk^-T (row per lane, regs)
    for (int i = k + 1; i < 4; ++i) {
      const int bi = 16 * i;
      if (l < 16) {
        float row[16];
#pragma unroll
        for (int j = 0; j < 16; ++j) row[j] = S[(bi + l) * SLDA + (bk + j)];
#pragma unroll
        for (int j = 0; j < 16; ++j) {
#pragma unroll
          for (int p = 0; p < j; ++p)
            row[j] -= row[p] * S[(bk + j) * SLDA + (bk + p)];
          row[j] /= S[(bk + j) * SLDA + (bk + j)];
        }
#pragma unroll
        for (int j = 0; j < 16; ++j) S[(bi + l) * SLDA + (bk + j)] = row[j];
      }
      __syncthreads();
    }

    // (3) Schur updates S_ij -= L_ik * L_jk^T via V_WMMA_F32_16X16X4_F32.
    //     A loaded negated so D = C + (-A)B^T (FP32 NEG is C-only per ISA).
    for (int i = k + 1; i < 4; ++i) {
      for (int j = k + 1; j <= i; ++j) {
        const int bi = 16 * i, bj = 16 * j;
        v8f c;
#pragma unroll
        for (int v = 0; v < 8; ++v)               // C/D layout: VGPR v -> M = v + 8h
          c[v] = S[(bi + v + 8 * h) * SLDA + (bj + m)];
#pragma unroll
        for (int kc = 0; kc < 4; ++kc) {          // K=16 as 4 chained K=4 WMMAs
          const int k0 = bk + 4 * kc + 2 * h;
          v2f a, b;
          a.x = -S[(bi + m) * SLDA + k0];         // A[m][2h], A[m][2h+1]
          a.y = -S[(bi + m) * SLDA + k0 + 1];
          b.x =  S[(bj + m) * SLDA + k0];         // B[2h][n]=L_jk[n][2h]
          b.y =  S[(bj + m) * SLDA + k0 + 1];
          c = __builtin_amdgcn_wmma_f32_16x16x4_f32(
              false, a, false, b, (short)0, c, false, false);
        }
        __syncthreads();
#pragma unroll
        for (int v = 0; v < 8; ++v)
          S[(bi + v + 8 * h) * SLDA + (bj + m)] = c[v];
        __syncthreads();
      }
    }
  }

  // ---- forward solve L y = (x - mu); lane l owns y[l], y[l+32] ----
  const float* xb = x + (size_t)pid * 64;
  const float* mb = mu + (size_t)pid * 64;
  float y0 = xb[l] - mb[l];
  float y1 = xb[l + 32] - mb[l + 32];
  for (int j = 0; j < 32; ++j) {
    float dj = S[j * SLDA + j];
    float yj = __shfl(y0, j, 32) / dj;
    if (l == j) y0 = yj;
    if (l > j) y0 -= S[l * SLDA + j] * yj;
    y1 -= S[(l + 32) * SLDA + j] * yj;
  }
  for (int j = 32; j < 64; ++j) {
    float dj = S[j * SLDA + j];
    float yj = __shfl(y1, j - 32, 32) / dj;
    if (l == j - 32) y1 = yj;
    else if (l > j - 32) y1 -= S[(l + 32) * SLDA + j] * yj;
  }

  // ---- quad + logdet wave reduction ----
  float quad = y0 * y0 + y1 * y1;
  float lg = logf(S[l * SLDA + l]) + logf(S[(l + 32) * SLDA + (l + 32)]);
#pragma unroll
  for (int off = 16; off > 0; off >>= 1) {
    quad += __shfl_xor(quad, off, 32);
    lg   += __shfl_xor(lg, off, 32);
  }
  if (l == 0)
    ll_out[pid] = -0.5f * (64.0f * LOG2PI + 2.0f * lg + quad);
}

// One wave32 per block; two (b,t) problems, async-double-buffered sigma tiles.
__global__ __launch_bounds__(32) void chol_ll_kernel(const float* __restrict__ x,
                                                     const float* __restrict__ mu,
                                                     const float* __restrict__ sigma,
                                                     float* __restrict__ ll_out) {
  __shared__ float S[PPB][LBUF];
  const int l = threadIdx.x;
  const int p0 = blockIdx.x * PPB;

  // Issue both tiles' async loads back-to-back (32 + 32 instructions).
  async_load_sigma(S[0], sigma + (size_t)p0 * 4096, l);
  async_load_sigma(S[1], sigma + (size_t)(p0 + 1) * 4096, l);

  // Async loads complete in order: <=32 outstanding  =>  buffer 0 resident,
  // while buffer 1's loads keep streaming behind the first factorization.
  asm volatile("s_wait_asynccnt 0x20" ::: "memory");
  __syncthreads();
  chol_ll_one(S[0], x, mu, p0, ll_out, l);

  asm volatile("s_wait_asynccnt 0x0" ::: "memory");
  __syncthreads();
  chol_ll_one(S[1], x, mu, p0 + 1, ll_out, l);
}

// Deterministic single-block tree reduction: nll = -(sum ll)/B
__global__ __launch_bounds__(256) void nll_reduce_kernel(const float* __restrict__ part,
                                                         float* __restrict__ out, int n) {
  __shared__ float sm[256];
  int tid = threadIdx.x;
  float s = 0.0f;
  for (int i = tid; i < n; i += 256) s += part[i];
  sm[tid] = s;
  __syncthreads();
  for (int off = 128; off > 0; off >>= 1) {
    if (tid < off) sm[tid] += sm[tid + off];
    __syncthreads();
  }
  if (tid == 0) out[0] = -sm[0] * (1.0f / 64.0f);
}

extern "C" void kernel_launch(void* const* d_in, const int* in_sizes, int n_in,
                              void* d_out, int out_size, void* d_ws, size_t ws_size,
                              hipStream_t stream) {
  const float* x     = (const float*)d_in[0];
  const float* mu    = (const float*)d_in[1];
  const float* sigma = (const float*)d_in[2];
  float* ll = (float*)d_ws;           // 25600 floats = 100KB scratch
  chol_ll_kernel<<<NPROB / PPB, 32, 0, stream>>>(x, mu, sigma, ll);
  nll_reduce_kernel<<<1, 256, 0, stream>>>(ll, (float*)d_out, NPROB);
}